// GeraclCore_54451595379274
// MI455X (gfx1250) — compile-verified
//
#include <hip/hip_runtime.h>
#include <hip/hip_bf16.h>
#include <math.h>

// Problem constants (from reference)
#define B_   16
#define S_   4096
#define H_   768
#define F_   2048
#define C_   8
#define NROW 144        // 128 class rows (b*C+c) + 16 text rows (128+b)
#define NTILE (NROW/16) // 9 row tiles; tile 8 = text tile

typedef __attribute__((ext_vector_type(2))) float v2f;
typedef __attribute__((ext_vector_type(8))) float v8f;

__device__ __forceinline__ v8f wmma_f32_16x16x4(v2f a, v2f b, v8f c) {
    // (neg_a, A, neg_b, B, c_mod, C, reuse_a, reuse_b)
    return __builtin_amdgcn_wmma_f32_16x16x4_f32(false, a, false, b, (short)0, c,
                                                 false, false);
}

// ---------------- kernel 0: zero accumulators ----------------
__global__ __launch_bounds__(256) void k_zero(float* p, int n) {
    int i = blockIdx.x * 256 + threadIdx.x;
    if (i < n) p[i] = 0.0f;
}

// ---------------- kernel 1: masked streaming reduction ----------------
// grid: (hChunk=3, sChunk=16, b=16), block 256 threads (one h each).
// accum rows: r<128 -> class sum for (b=r/8, c=r%8); r>=128 -> text sum for b=r-128.
__global__ __launch_bounds__(256) void k_reduce(const float* __restrict__ emb,
                                                const int* __restrict__ mask,
                                                float* __restrict__ accum,
                                                float* __restrict__ counts) {
    const int hChunk = blockIdx.x;          // 0..2
    const int sChunk = blockIdx.y;          // 0..15
    const int b      = blockIdx.z;          // 0..15
    const int h      = hChunk * 256 + threadIdx.x;
    const int s0     = sChunk * (S_ / 16);

    const float* eb = emb + (size_t)b * S_ * H_;
    const int*   mb = mask + (size_t)b * S_;

    float accT = 0.0f;
    float accC[C_];
#pragma unroll
    for (int c = 0; c < C_; ++c) accC[c] = 0.0f;

    for (int s = s0; s < s0 + (S_ / 16); ++s) {
        int m = mb[s];                       // wave-uniform -> uniform branch
        if (m == -3) {
            accT += eb[(size_t)s * H_ + h];
        } else if (m >= 0 && m < C_) {
            float v = eb[(size_t)s * H_ + h];
#pragma unroll
            for (int c = 0; c < C_; ++c)
                if (m == c) accC[c] += v;
        }
        // separators (-1,-2) and pad (-5): skip load entirely
    }

    if (accT != 0.0f) atomicAdd(&accum[(size_t)(128 + b) * H_ + h], accT);
#pragma unroll
    for (int c = 0; c < C_; ++c)
        if (accC[c] != 0.0f) atomicAdd(&accum[(size_t)(b * C_ + c) * H_ + h], accC[c]);

    // counts: one thread per (b, sChunk) scans its range once
    if (hChunk == 0 && threadIdx.x == 0) {
        float cT = 0.0f;
        float cC[C_];
#pragma unroll
        for (int c = 0; c < C_; ++c) cC[c] = 0.0f;
        for (int s = s0; s < s0 + (S_ / 16); ++s) {
            int m = mb[s];
            if (m == -3) cT += 1.0f;
            else if (m >= 0 && m < C_) {
#pragma unroll
                for (int c = 0; c < C_; ++c)
                    if (m == c) cC[c] += 1.0f;
            }
        }
        if (cT != 0.0f) atomicAdd(&counts[128 + b], cT);
#pragma unroll
        for (int c = 0; c < C_; ++c)
            if (cC[c] != 0.0f) atomicAdd(&counts[b * C_ + c], cC[c]);
    }
}

// ---------------- kernel 2: in-place divide (sums -> means) ----------------
__global__ __launch_bounds__(256) void k_mean(float* __restrict__ accum,
                                              const float* __restrict__ counts) {
    int idx = blockIdx.x * 256 + threadIdx.x;
    if (idx >= NROW * H_) return;
    int r = idx / H_;
    accum[idx] = accum[idx] / fmaxf(counts[r], 1e-9f);
}

// ---------------- kernel 3: GEMM1 + bias + exact GELU ----------------
// X(144,768) @ W1(768,2048) + b1 -> gelu -> Y(144,2048)
// One wave per 16x16 tile; K-loop 768/4 = 192 v_wmma_f32_16x16x4_f32 ops.
__global__ __launch_bounds__(32) void k_gemm1(const float* __restrict__ X,
                                              const float* __restrict__ W1c,
                                              const float* __restrict__ b1c,
                                              const float* __restrict__ W1t,
                                              const float* __restrict__ b1t,
                                              float* __restrict__ Y) {
    const int rTile   = blockIdx.y;            // 0..8 (8 = text)
    const int colBase = blockIdx.x * 16;       // 0..2047
    const float* W    = (rTile == 8) ? W1t : W1c;
    const float* bias = (rTile == 8) ? b1t : b1c;

    const int lane = threadIdx.x;              // 0..31
    const int mrow = lane & 15;                // A row within tile
    const int koff = (lane >> 4) << 1;         // 0 (lanes 0-15) / 2 (lanes 16-31)
    const int col  = colBase + (lane & 15);
    const float* arow = X + (size_t)(rTile * 16 + mrow) * H_;

    v8f acc = {};
    for (int k0 = 0; k0 < H_; k0 += 4) {
        v2f a = *(const v2f*)(arow + k0 + koff);           // A: K=koff,koff+1
        v2f b;
        b.x = W[(size_t)(k0 + koff) * F_ + col];           // B: K=koff  row
        b.y = W[(size_t)(k0 + koff + 1) * F_ + col];       // B: K=koff+1 row
        acc = wmma_f32_16x16x4(a, b, acc);
    }

    const float bv = bias[col];
#pragma unroll
    for (int j = 0; j < 8; ++j) {                          // D: M = j + 8*(lane>=16)
        int row = rTile * 16 + j + ((lane >> 4) << 3);
        float x = acc[j] + bv;
        float g = 0.5f * x * (1.0f + erff(x * 0.70710678118654752f));
        Y[(size_t)row * F_ + col] = g;
    }
}

// ---------------- kernel 4: GEMM2 + bias ----------------
// Y(144,2048) @ W2(2048,768) + b2 -> Z(144,768)
__global__ __launch_bounds__(32) void k_gemm2(const float* __restrict__ Y,
                                              const float* __restrict__ W2c,
                                              const float* __restrict__ b2c,
                                              const float* __restrict__ W2t,
                                              const float* __restrict__ b2t,
                                              float* __restrict__ Z) {
    const int rTile   = blockIdx.y;            // 0..8
    const int colBase = blockIdx.x * 16;       // 0..767
    const float* W    = (rTile == 8) ? W2t : W2c;
    const float* bias = (rTile == 8) ? b2t : b2c;

    const int lane = threadIdx.x;
    const int mrow = lane & 15;
    const int koff = (lane >> 4) << 1;
    const int col  = colBase + (lane & 15);
    const float* arow = Y + (size_t)(rTile * 16 + mrow) * F_;

    v8f acc = {};
    for (int k0 = 0; k0 < F_; k0 += 4) {
        v2f a = *(const v2f*)(arow + k0 + koff);
        v2f b;
        b.x = W[(size_t)(k0 + koff) * H_ + col];
        b.y = W[(size_t)(k0 + koff + 1) * H_ + col];
        acc = wmma_f32_16x16x4(a, b, acc);
    }

    const float bv = bias[col];
#pragma unroll
    for (int j = 0; j < 8; ++j) {
        int row = rTile * 16 + j + ((lane >> 4) << 3);
        Z[(size_t)row * H_ + col] = acc[j] + bv;
    }
}

// ---------------- kernel 5: per-pair dot products ----------------
// out[b*C+c] = dot(Z[b*C+c,:], Z[128+b,:]) over H
__global__ __launch_bounds__(128) void k_sim(const float* __restrict__ Z,
                                             float* __restrict__ out) {
    int i = blockIdx.x * 4 + (threadIdx.x >> 5);   // pair index 0..127
    int lane = threadIdx.x & 31;
    if (i >= B_ * C_) return;
    int b = i >> 3;
    const float* zc = Z + (size_t)i * H_;
    const float* zt = Z + (size_t)(128 + b) * H_;
    float acc = 0.0f;
    for (int h = lane; h < H_; h += 32) acc += zc[h] * zt[h];
#pragma unroll
    for (int off = 16; off > 0; off >>= 1) acc += __shfl_xor(acc, off, 32);
    if (lane == 0) out[i] = acc;
}

extern "C" void kernel_launch(void* const* d_in, const int* in_sizes, int n_in,
                              void* d_out, int out_size, void* d_ws, size_t ws_size,
                              hipStream_t stream) {
    (void)in_sizes; (void)n_in; (void)out_size; (void)ws_size;
    const float* emb  = (const float*)d_in[0];
    const int*   mask = (const int*)d_in[1];   // classes_mask (int)
    // d_in[2] = classes_count (uniformly C, static in reference)
    const float* W1t = (const float*)d_in[3];
    const float* b1t = (const float*)d_in[4];
    const float* W2t = (const float*)d_in[5];
    const float* b2t = (const float*)d_in[6];
    const float* W1c = (const float*)d_in[7];
    const float* b1c = (const float*)d_in[8];
    const float* W2c = (const float*)d_in[9];
    const float* b2c = (const float*)d_in[10];
    float* out = (float*)d_out;

    // workspace layout (floats)
    float* ws     = (float*)d_ws;
    float* accum  = ws;                        // NROW*H  (sums, then means == X)
    float* counts = accum + NROW * H_;         // 256 (padded)
    float* Y      = counts + 256;              // NROW*F
    float* Z      = Y + (size_t)NROW * F_;     // NROW*H

    const int nzero = NROW * H_ + 256;
    k_zero<<<dim3((nzero + 255) / 256), dim3(256), 0, stream>>>(accum, nzero);

    k_reduce<<<dim3(3, 16, 16), dim3(256), 0, stream>>>(emb, mask, accum, counts);

    k_mean<<<dim3((NROW * H_ + 255) / 256), dim3(256), 0, stream>>>(accum, counts);

    k_gemm1<<<dim3(F_ / 16, NTILE), dim3(32), 0, stream>>>(accum, W1c, b1c, W1t, b1t, Y);

    k_gemm2<<<dim3(H_ / 16, NTILE), dim3(32), 0, stream>>>(Y, W2c, b2c, W2t, b2t, Z);

    k_sim<<<dim3((B_ * C_) / 4), dim3(128), 0, stream>>>(Z, out);
}